// PowerSpectrum_72181220376899
// MI455X (gfx1250) — compile-verified
//
#include <hip/hip_runtime.h>
#include <math.h>

// CDNA5 / gfx1250, wave32.
typedef __attribute__((ext_vector_type(2))) float v2f;
typedef __attribute__((ext_vector_type(8))) float v8f;

#define ACH     64      // A = nsp * nmax
#define PS_L    2080    // 64 diag + 2016 off-diag per degree
#define DTOT    14560   // 7 * 2080
#define NTHREAD 256
#define NWAVE   8
#define NJOBS   70      // 7 degrees * 10 upper-triangle 16x16 tiles
#define SE_TOT  4096    // 64 * sum(Kpad) = 64 * (4+4+8+8+12+12+16)

__global__ __launch_bounds__(NTHREAD) void ps_wmma_kernel(
    const float* __restrict__ se0, const float* __restrict__ se1,
    const float* __restrict__ se2, const float* __restrict__ se3,
    const float* __restrict__ se4, const float* __restrict__ se5,
    const float* __restrict__ se6, float* __restrict__ out)
{
    __shared__ __align__(16) float lds_ps[DTOT];   // staged powerspectrum (58 KB)
    __shared__ __align__(16) float lds_se[SE_TOT]; // zero-padded se rows   (16 KB)
    __shared__ float red[NWAVE];
    __shared__ float scale_sh;

    const int env  = blockIdx.x;
    const int tid  = threadIdx.x;
    const int wave = __builtin_amdgcn_readfirstlane(tid >> 5);  // wave-uniform
    const int lane = tid & 31;
    const int lm   = lane & 15;
    const int half = lane >> 4;        // 0: K pair {0,1}; 1: K pair {2,3}

    // ---- Phase 0: stage se_l into LDS, rows zero-padded to Kpad --------
    {
        const float* seArr[7] = {se0, se1, se2, se3, se4, se5, se6};
        const int baseTab[7]  = {0, 256, 512, 1024, 1536, 2304, 3072};
#pragma unroll
        for (int l = 0; l < 7; ++l) {            // l compile-time after unroll
            const int K  = 2 * l + 1;
            const int KP = (K + 3) & ~3;
            const float* S = seArr[l] + (size_t)env * (ACH * K);
            for (int idx = tid; idx < ACH * KP; idx += NTHREAD) {
                const int row = idx / KP;        // KP constant -> mul/shift
                const int kk  = idx - row * KP;
                // branchless: clamped-address load + value select
                const float v = S[row * K + (kk < K ? kk : 0)];
                lds_se[baseTab[l] + idx] = (kk < K) ? v : 0.f;
            }
        }
    }
    __syncthreads();

    // ---- Phase 1: 70 wave-level WMMA tile jobs -------------------------
    for (int job = wave; job < NJOBS; job += NWAVE) {
        const int l = job / 10;
        const int t = job - l * 10;          // upper-triangle tile id 0..9
        const int K  = 2 * l + 1;
        const int KP = (K + 3) & ~3;

        // map t -> (ti, tj), ti <= tj
        const int ti   = (t < 4) ? 0 : (t < 7) ? 1 : (t < 9) ? 2 : 3;
        const int tb   = (t < 4) ? 0 : (t < 7) ? 4 : (t < 9) ? 7 : 9;
        const int tj   = ti + (t - tb);

        int seBase;
        switch (l) {                          // scalar (l is wave-uniform)
            case 0: seBase = 0;    break;
            case 1: seBase = 256;  break;
            case 2: seBase = 512;  break;
            case 3: seBase = 1024; break;
            case 4: seBase = 1536; break;
            case 5: seBase = 2304; break;
            default: seBase = 3072; break;
        }
        const float* Sl = &lds_se[seBase];

        const int rowA = ti * 16 + lm;       // A fragment: M = lm
        const int rowB = tj * 16 + lm;       // B fragment: N = lm (mirrored layout)
        const int kb   = half * 2;

        v8f c = {0.f, 0.f, 0.f, 0.f, 0.f, 0.f, 0.f, 0.f};
        const int steps = KP >> 2;
        for (int s = 0; s < steps; ++s) {
            const int k0 = 4 * s + kb;
            // unconditional 8B-aligned LDS loads (rows padded to 16B multiples)
            const v2f a = *(const v2f*)&Sl[rowA * KP + k0];
            const v2f b = *(const v2f*)&Sl[rowB * KP + k0];
            // D = A * B + C, fp32 exact (matches reference precision)
            c = __builtin_amdgcn_wmma_f32_16x16x4_f32(
                    false, a, false, b, (short)0, c, false, false);
        }

        // scatter C fragment into LDS in powerspectrum order
        const float invs = 1.f / sqrtf((float)K);            // 1/sqrt(2l+1)
        const float offs = invs * 1.41421356237309504880f;   // sqrt2/sqrt(2l+1)
        const int q     = tj * 16 + lm;                      // column index
        const int mbase = ti * 16 + half * 8;                // C layout: M = r (+8)
        const int lbase = l * PS_L;
        if (ti == tj) {                       // diagonal tile: conditional scatter
#pragma unroll
            for (int r = 0; r < 8; ++r) {
                const int p = mbase + r;
                if (p == q) {
                    lds_ps[lbase + p] = c[r] * invs;
                } else if (p < q) {
                    const int idx = 64 + 63 * p - ((p * (p - 1)) >> 1) + (q - p - 1);
                    lds_ps[lbase + idx] = c[r] * offs;
                }
            }
        } else {                              // off-diagonal tile: p < q always
#pragma unroll
            for (int r = 0; r < 8; ++r) {
                const int p = mbase + r;
                const int idx = 64 + 63 * p - ((p * (p - 1)) >> 1) + (q - p - 1);
                lds_ps[lbase + idx] = c[r] * offs;
            }
        }
    }
    __syncthreads();

    // ---- Phase 2: L2 norm over all 14560 entries -----------------------
    float ss = 0.f;
    {
        const float4* p4 = (const float4*)lds_ps;
        for (int idx = tid; idx < DTOT / 4; idx += NTHREAD) {
            const float4 v = p4[idx];
            ss += v.x * v.x + v.y * v.y + v.z * v.z + v.w * v.w;
        }
    }
#pragma unroll
    for (int off = 16; off > 0; off >>= 1)
        ss += __shfl_down(ss, off, 32);
    if (lane == 0) red[wave] = ss;
    __syncthreads();
    if (tid == 0) {
        float tot = 0.f;
#pragma unroll
        for (int w = 0; w < NWAVE; ++w) tot += red[w];
        scale_sh = 1.f / fmaxf(sqrtf(tot), 1e-12f);
    }
    __syncthreads();
    const float scale = scale_sh;

    // ---- Phase 3: coalesced 128-bit streaming writeout -----------------
    float4*       o4 = (float4*)(out + (size_t)env * DTOT);
    const float4* p4 = (const float4*)lds_ps;
    for (int idx = tid; idx < DTOT / 4; idx += NTHREAD) {
        float4 v = p4[idx];
        v.x *= scale; v.y *= scale; v.z *= scale; v.w *= scale;
        o4[idx] = v;
    }
}

extern "C" void kernel_launch(void* const* d_in, const int* in_sizes, int n_in,
                              void* d_out, int out_size, void* d_ws, size_t ws_size,
                              hipStream_t stream) {
    const float* se0 = (const float*)d_in[0];
    const float* se1 = (const float*)d_in[1];
    const float* se2 = (const float*)d_in[2];
    const float* se3 = (const float*)d_in[3];
    const float* se4 = (const float*)d_in[4];
    const float* se5 = (const float*)d_in[5];
    const float* se6 = (const float*)d_in[6];
    float* out = (float*)d_out;

    const int J = in_sizes[0] / (ACH * 1);   // se0 is [J, 64, 1]
    ps_wmma_kernel<<<J, NTHREAD, 0, stream>>>(se0, se1, se2, se3, se4, se5, se6, out);
}